// ManifoldEmbLoss_29257317220640
// MI455X (gfx1250) — compile-verified
//
#include <hip/hip_runtime.h>
#include <hip/hip_bf16.h>

typedef __attribute__((ext_vector_type(2))) float v2f;
typedef __attribute__((ext_vector_type(8))) float v8f;

#define N_PTS 8192
#define XD 64
#define MD 16
#define KNN 10
#define KP1 11
#define ROWS_PER_WG 16
#define COLS_PER_TILE 128
#define NUM_TILES (N_PTS / COLS_PER_TILE)   // 64
#define NBLOCKS (N_PTS / ROWS_PER_WG)       // 512
#define BIGF 3.0e38f

// ---------------- kernel 1: squared norms of X rows ----------------
__global__ __launch_bounds__(256) void sqnorm_kernel(const float* __restrict__ X,
                                                     float* __restrict__ sq) {
    int i = blockIdx.x * blockDim.x + threadIdx.x;
    if (i < N_PTS) {
        float s = 0.0f;
        #pragma unroll
        for (int d = 0; d < XD; ++d) {
            float v = X[i * XD + d];
            s += v * v;
        }
        sq[i] = s;
    }
}

// ---------------- kernel 2: WMMA Gram tiles + top-11 + per-row loss ----------------
__global__ __launch_bounds__(256) void knn_loss_kernel(const float* __restrict__ z,
                                                       const float* __restrict__ X,
                                                       const float* __restrict__ sq,
                                                       float* __restrict__ partials) {
    __shared__ float lds_d2[ROWS_PER_WG * COLS_PER_TILE];   // 8 KB
    __shared__ float lds_bd[256 * KP1];                     // 11 KB
    __shared__ int   lds_bi[256 * KP1];                     // 11 KB
    __shared__ float lds_rowsum[ROWS_PER_WG];

    const int tid  = threadIdx.x;
    const int wave = tid >> 5;
    const int lane = tid & 31;
    const int l16  = lane & 15;
    const int hi   = lane >> 4;            // 0 or 1 (half-wave)
    const int rowBase = blockIdx.x * ROWS_PER_WG;

    // A fragments (query rows), register-resident for all column tiles.
    // f32 16x4 A layout: lane m<16 holds {A[m][0],A[m][1]}, lane m+16 holds {A[m][2],A[m][3]}
    v2f afrag[16];
    {
        const float* xq = X + (rowBase + l16) * XD + 2 * hi;
        #pragma unroll
        for (int k = 0; k < 16; ++k)
            afrag[k] = *(const v2f*)(xq + 4 * k);
    }

    // sq of rows this lane owns in the C/D layout: VGPR v holds M = v + 8*hi
    float sqm[8];
    #pragma unroll
    for (int v = 0; v < 8; ++v)
        sqm[v] = sq[rowBase + v + 8 * hi];

    // per-thread partial top-11 (sorted ascending), register-resident
    float bd[KP1]; int bi[KP1];
    #pragma unroll
    for (int j = 0; j < KP1; ++j) { bd[j] = BIGF; bi[j] = 0; }
    float worst = BIGF;

    const int t_row = tid >> 4;   // 0..15 : row within block
    const int t_seg = tid & 15;   // 0..15 : 8-column slice within 128-tile

    for (int tile = 0; tile < NUM_TILES; ++tile) {
        const int c = tile * COLS_PER_TILE + wave * 16 + l16;   // column point
        const float* xc = X + c * XD + 2 * hi;
        const float sqc = sq[c];

        v8f acc = {0.f, 0.f, 0.f, 0.f, 0.f, 0.f, 0.f, 0.f};
        #pragma unroll
        for (int k = 0; k < 16; ++k) {
            v2f bfrag = *(const v2f*)(xc + 4 * k);
            acc = __builtin_amdgcn_wmma_f32_16x16x4_f32(
                      false, afrag[k], false, bfrag,
                      (short)0, acc, false, false);
        }

        // expand dot -> squared distance, store 16x16 sub-tile to LDS
        #pragma unroll
        for (int v = 0; v < 8; ++v) {
            float d2 = sqm[v] + sqc - 2.0f * acc[v];
            lds_d2[(v + 8 * hi) * COLS_PER_TILE + wave * 16 + l16] = d2;
        }
        __syncthreads();

        // selection: every thread scans its 8-column slice of its row
        const int cbase = tile * COLS_PER_TILE + t_seg * 8;
        #pragma unroll
        for (int j = 0; j < 8; ++j) {
            float d = lds_d2[t_row * COLS_PER_TILE + t_seg * 8 + j];
            if (d < worst) {
                float dc = d; int ic = cbase + j;
                #pragma unroll
                for (int s = 0; s < KP1; ++s) {
                    if (dc < bd[s]) {
                        float td = bd[s]; bd[s] = dc; dc = td;
                        int   ti = bi[s]; bi[s] = ic; ic = ti;
                    }
                }
                worst = bd[KP1 - 1];
            }
        }
        __syncthreads();
    }

    // publish partial lists
    #pragma unroll
    for (int j = 0; j < KP1; ++j) {
        lds_bd[tid * KP1 + j] = bd[j];
        lds_bi[tid * KP1 + j] = bi[j];
    }
    __syncthreads();

    // one thread per row merges the 16 partial lists, then computes the row loss
    if (t_seg == 0) {
        float fd[KP1]; int fi[KP1];
        #pragma unroll
        for (int j = 0; j < KP1; ++j) { fd[j] = bd[j]; fi[j] = bi[j]; }
        float w = fd[KP1 - 1];
        for (int p = 1; p < 16; ++p) {
            const int base = (t_row * 16 + p) * KP1;
            #pragma unroll
            for (int j = 0; j < KP1; ++j) {
                float d = lds_bd[base + j];
                if (d < w) {
                    float dc = d; int ic = lds_bi[base + j];
                    #pragma unroll
                    for (int s = 0; s < KP1; ++s) {
                        if (dc < fd[s]) {
                            float td = fd[s]; fd[s] = dc; dc = td;
                            int   ti = fi[s]; fi[s] = ic; ic = ti;
                        }
                    }
                    w = fd[KP1 - 1];
                }
            }
        }

        // fd/fi sorted ascending; slot 0 is self -> neighbors are fi[1..10]
        const int r = rowBase + t_row;
        float xd[KNN], zd[KNN];
        #pragma unroll 2
        for (int j = 0; j < KNN; ++j) {
            const int nb = fi[j + 1];
            float sx = 0.0f;
            for (int d = 0; d < XD; ++d) {
                float df = X[r * XD + d] - X[nb * XD + d];
                sx += df * df;
            }
            xd[j] = sqrtf(sx);
            float sz = 0.0f;
            for (int d = 0; d < MD; ++d) {
                float df = z[r * MD + d] - z[nb * MD + d];
                sz += df * df;
            }
            zd[j] = sqrtf(sz);
        }
        float xmax = 0.0f, zmax = 0.0f;
        #pragma unroll
        for (int j = 0; j < KNN; ++j) {
            xmax = fmaxf(xmax, xd[j]);
            zmax = fmaxf(zmax, zd[j]);
        }
        xmax = fmaxf(xmax, 1e-8f);
        zmax = fmaxf(zmax, 1e-8f);
        float rsum = 0.0f;
        #pragma unroll
        for (int j = 0; j < KNN; ++j)
            rsum += fabsf(zd[j] / zmax - xd[j] / xmax);
        lds_rowsum[t_row] = rsum;
    }
    __syncthreads();

    if (tid == 0) {
        float s = 0.0f;
        #pragma unroll
        for (int i = 0; i < ROWS_PER_WG; ++i) s += lds_rowsum[i];
        partials[blockIdx.x] = s;
    }
}

// ---------------- kernel 3: deterministic final reduction ----------------
__global__ void final_reduce_kernel(const float* __restrict__ partials,
                                    float* __restrict__ out) {
    if (threadIdx.x == 0 && blockIdx.x == 0) {
        float s = 0.0f;
        for (int i = 0; i < NBLOCKS; ++i) s += partials[i];
        out[0] = s / (float)(N_PTS * KNN);
    }
}

extern "C" void kernel_launch(void* const* d_in, const int* in_sizes, int n_in,
                              void* d_out, int out_size, void* d_ws, size_t ws_size,
                              hipStream_t stream) {
    const float* z = (const float*)d_in[0];   // [8192, 16]
    const float* X = (const float*)d_in[1];   // [8192, 64]
    float* out = (float*)d_out;               // scalar

    float* sq       = (float*)d_ws;                    // 8192 floats
    float* partials = (float*)d_ws + N_PTS;            // 512 floats

    sqnorm_kernel<<<N_PTS / 256, 256, 0, stream>>>(X, sq);
    knn_loss_kernel<<<NBLOCKS, 256, 0, stream>>>(z, X, sq, partials);
    final_reduce_kernel<<<1, 32, 0, stream>>>(partials, out);
}